// FirstHitProjector_77025943486550
// MI455X (gfx1250) — compile-verified
//
#include <hip/hip_runtime.h>

// Problem constants from the reference: (B=8, E=2, K=64, H=256, W=256)
#define FHP_B   8
#define FHP_K   64
#define FHP_HW  65536           // H*W
#define FHP_EPS 1e-6f
#define STAGES  4               // async pipeline depth (power of two)

// ---- gfx1250 async global->LDS path probe -------------------------------
#if defined(__AMDGCN__) &&                                                   \
    __has_builtin(__builtin_amdgcn_global_load_async_to_lds_b128) &&         \
    __has_builtin(__builtin_amdgcn_s_wait_asynccnt)
#define FHP_ASYNC 1
#else
#define FHP_ASYNC 0
#endif

#if FHP_ASYNC
// Builtin signature (from hipcc diagnostic): first param is
// 'int __vector(4) __device__ *' i.e. ptr to <4 x i32> in addrspace(1);
// second is the LDS-side pointer (addrspace(3)), then imm offset, imm cpol.
typedef int fhp_v4i __attribute__((ext_vector_type(4)));
typedef __attribute__((address_space(1))) fhp_v4i fhp_g4;
typedef __attribute__((address_space(3))) fhp_v4i fhp_l4;

// Integer round-trip casts: generic->AS1 is the identity VA; generic LDS
// address low 32 bits are the wave-relative LDS offset (aperture mapping),
// which is exactly what the async instruction's VDST VGPR wants.
__device__ __forceinline__ void fhp_async_b128(const void* g, void* l) {
  __builtin_amdgcn_global_load_async_to_lds_b128(
      (fhp_g4*)(unsigned long long)(size_t)g,
      (fhp_l4*)(unsigned)(size_t)l,
      0, 0);
}
#endif

// One scan step for one pixel.  Order matters: trans_before[k] (= r before
// update) feeds front/back accumulation; r is updated last.
__device__ __forceinline__ void fhp_step(float p0, float p1,
                                         float& r, float& f0, float& f1,
                                         float& bk0, float& bk1,
                                         float& q0, float& q1, float& vc) {
  f0 = fmaf(r, p0, f0);                 // visible/front expectation, e0
  f1 = fmaf(r, p1, f1);                 // visible/front expectation, e1
  const float hb = 1.0f - r;            // has_front_before
  bk0 = fmaf(hb, p0, bk0);
  bk1 = fmaf(hb, p1, bk1);
  const float m   = fmaxf(p0, p1);
  const float cls = (m > 0.5f) ? ((p1 > p0) ? 2.0f : 1.0f) : 0.0f;
  vc = (vc == 0.0f) ? cls : vc;         // first non-background class wins
  const float n0 = 1.0f - p0;
  const float n1 = 1.0f - p1;
  q0 *= n0;                             // amodal products
  q1 *= n1;
  r  *= n0 * n1;                        // cumprod of (1 - occ)
}

__global__ __launch_bounds__(256) void fhp_kernel(const float* __restrict__ in,
                                                  float* __restrict__ out) {
#if FHP_ASYNC
  __shared__ float4 stage[STAGES][2][256];   // [slot][entity][thread], 32 KB
#endif
  const int tid = blockIdx.x * 256 + (int)threadIdx.x;
  const int p   = tid << 2;                  // first of 4 pixels (float4 lane)
  if (p >= FHP_B * FHP_HW) return;
  const int b  = p >> 16;                    // p / HW
  const int hw = p & (FHP_HW - 1);           // p % HW  (HW=65536, 4-divisible)

  // entity_probs[b][e][k][hw] : element ((b*2+e)*64 + k)*HW + hw
  const float* pe0 = in + (size_t)(b * 2 * FHP_K) * FHP_HW + hw;
  const float* pe1 = pe0 + (size_t)FHP_K * FHP_HW;

  float r[4], f0[4], f1[4], bk0[4], bk1[4], q0[4], q1[4], vc[4];
#pragma unroll
  for (int j = 0; j < 4; ++j) {
    r[j] = 1.0f; f0[j] = f1[j] = bk0[j] = bk1[j] = 0.0f;
    q0[j] = q1[j] = 1.0f; vc[j] = 0.0f;
  }

#if FHP_ASYNC
  // Prologue: prime STAGES-1 slices.
#pragma unroll
  for (int k0 = 0; k0 < STAGES - 1; ++k0) {
    fhp_async_b128(pe0 + (size_t)k0 * FHP_HW, &stage[k0][0][threadIdx.x]);
    fhp_async_b128(pe1 + (size_t)k0 * FHP_HW, &stage[k0][1][threadIdx.x]);
  }
#endif

  for (int k = 0; k < FHP_K; ++k) {
    float4 v0, v1;
#if FHP_ASYNC
    const int kn = k + (STAGES - 1);
    if (kn < FHP_K) {
      const int sn = kn & (STAGES - 1);
      fhp_async_b128(pe0 + (size_t)kn * FHP_HW, &stage[sn][0][threadIdx.x]);
      fhp_async_b128(pe1 + (size_t)kn * FHP_HW, &stage[sn][1][threadIdx.x]);
      // async loads complete in order: <= 2*(STAGES-1) outstanding
      // guarantees slice k has fully landed in LDS.
      __builtin_amdgcn_s_wait_asynccnt(2 * (STAGES - 1));
    } else {
      __builtin_amdgcn_s_wait_asynccnt(0);   // drain tail
    }
    const int s = k & (STAGES - 1);
    v0 = stage[s][0][threadIdx.x];
    v1 = stage[s][1][threadIdx.x];
#else
    v0 = *(const float4*)(pe0 + (size_t)k * FHP_HW);
    v1 = *(const float4*)(pe1 + (size_t)k * FHP_HW);
#if defined(__AMDGCN__)
    if (k + 4 < FHP_K) {
      __builtin_prefetch(pe0 + (size_t)(k + 4) * FHP_HW, 0, 0);
      __builtin_prefetch(pe1 + (size_t)(k + 4) * FHP_HW, 0, 0);
    }
#endif
#endif
    fhp_step(v0.x, v1.x, r[0], f0[0], f1[0], bk0[0], bk1[0], q0[0], q1[0], vc[0]);
    fhp_step(v0.y, v1.y, r[1], f0[1], f1[1], bk0[1], bk1[1], q0[1], q1[1], vc[1]);
    fhp_step(v0.z, v1.z, r[2], f0[2], f1[2], bk0[2], bk1[2], q0[2], q1[2], vc[2]);
    fhp_step(v0.w, v1.w, r[3], f0[3], f1[3], bk0[3], bk1[3], q0[3], q1[3], vc[3]);
  }

  // ---- Epilogue: per-pixel normalizations (match reference's arithmetic) --
  float st[3][4], bo[3][4], am0v[4], am1v[4], vcf[4];
#pragma unroll
  for (int j = 0; j < 4; ++j) {
    const float fs  = r[j] + f0[j] + f1[j];
    const float den = fmaxf(fs, FHP_EPS);          // clip(sum, EPS)
    const float fpb = r[j]  / den;
    const float fpa = f0[j] / den;
    const float fpc = f1[j] / den;
    const float h0 = (vc[j] == 0.0f) ? 1.0f : 0.0f;
    const float h1 = (vc[j] == 1.0f) ? 1.0f : 0.0f;
    const float h2 = (vc[j] == 2.0f) ? 1.0f : 0.0f;
    st[0][j] = (h0 - fpb) + fpb;                   // straight-through trick
    st[1][j] = (h1 - fpa) + fpa;
    st[2][j] = (h2 - fpc) + fpc;
    const float bs = bk0[j] + bk1[j];              // back channel 0 is 0
    if (bs > FHP_EPS) {
      bo[0][j] = 0.0f; bo[1][j] = bk0[j] / bs; bo[2][j] = bk1[j] / bs;
    } else {
      bo[0][j] = 0.0f; bo[1][j] = bk0[j];      bo[2][j] = bk1[j];
    }
    am0v[j] = 1.0f - q0[j];
    am1v[j] = 1.0f - q1[j];
    vcf[j]  = vc[j];
  }

  // ---- Stores: tuple outputs concatenated flat ----------------------------
  float* o_f = out + (size_t)(b * 3) * FHP_HW + hw;                      // front_probs_st (B,3,HW)
  float* o_b = out + (size_t)(3 * FHP_B) * FHP_HW
                   + (size_t)(b * 3) * FHP_HW + hw;                      // back_probs (B,3,HW)
  float* o_s = out + (size_t)(6 * FHP_B) * FHP_HW
                   + (size_t)b * FHP_HW + hw;                            // amodal_e0 base (B,HW)
#pragma unroll
  for (int c = 0; c < 3; ++c) {
    *(float4*)(o_f + (size_t)c * FHP_HW) =
        make_float4(st[c][0], st[c][1], st[c][2], st[c][3]);
    *(float4*)(o_b + (size_t)c * FHP_HW) =
        make_float4(bo[c][0], bo[c][1], bo[c][2], bo[c][3]);
  }
  *(float4*)(o_s)                                  = make_float4(am0v[0], am0v[1], am0v[2], am0v[3]);
  *(float4*)(o_s + (size_t)(1 * FHP_B) * FHP_HW)   = make_float4(am1v[0], am1v[1], am1v[2], am1v[3]);
  *(float4*)(o_s + (size_t)(2 * FHP_B) * FHP_HW)   = make_float4(f0[0], f0[1], f0[2], f0[3]);   // visible_e0
  *(float4*)(o_s + (size_t)(3 * FHP_B) * FHP_HW)   = make_float4(f1[0], f1[1], f1[2], f1[3]);   // visible_e1
  *(float4*)(o_s + (size_t)(4 * FHP_B) * FHP_HW)   = make_float4(vcf[0], vcf[1], vcf[2], vcf[3]); // visible_class
}

extern "C" void kernel_launch(void* const* d_in, const int* in_sizes, int n_in,
                              void* d_out, int out_size, void* d_ws, size_t ws_size,
                              hipStream_t stream) {
  (void)in_sizes; (void)n_in; (void)out_size; (void)d_ws; (void)ws_size;
  const float* in = (const float*)d_in[0];
  float* out = (float*)d_out;
  const int total_threads = (FHP_B * FHP_HW) / 4;   // 131072 threads, 4 px each
  dim3 block(256);
  dim3 grid(total_threads / 256);                   // 512 blocks
  hipLaunchKernelGGL(fhp_kernel, grid, block, 0, stream, in, out);
}